// SigmoidDiffWindow2D_32512902431515
// MI455X (gfx1250) — compile-verified
//
#include <hip/hip_runtime.h>
#include <math.h>

// SigmoidDiffWindow2D: out[k, j] = x[k, s+j] * window(k, s+j)
//   s  = round_half_even(start_idx[k])
//   window = 2*(g(kk*u1) - g(kk*u2)), g(z) = sigmoid(softplus(z)) = 1 - 1/(2+e^z)
//   u1 = (s+j) - (start_idx[k] + EPS), u2 = u1 - L
//
// Bandwidth-bound: 32 MB read (only the sliced region of x) + 32 MB write.
// No matmul structure -> WMMA is inapplicable; the optimal gfx1250 path is
// dense coalesced global loads/stores + v_exp_f32/v_rcp_f32 VALU.

#define T_DIM 16384
#define K_DIM 4096
#define L_DIM 2048
#define EPS_W 1e-6f
#define LOG2E 1.44269504088896340736f

__global__ __launch_bounds__(512) void sigdiff_window_slice_kernel(
    const float* __restrict__ x,          // (K, T)
    const float* __restrict__ k_param,    // (K,)
    const float* __restrict__ start_idx,  // (K,)
    float* __restrict__ out)              // (K, L)
{
    const int row = blockIdx.x;

    // Block-uniform row parameters -> scalar loads.
    const float kk = k_param[row];
    const float sf = start_idx[row];
    const int   si = (int)rintf(sf);      // v_rndne_f32: round-half-even, matches jnp.round

    // exp(kk*u) == exp2(c*u); fold start offset into per-row bases so the
    // inner loop is: e = exp2(c*j + d).
    const float c  = kk * LOG2E;
    const float b1 = (float)si - sf - EPS_W;        // u1 = j + b1
    const float b2 = b1 - (float)L_DIM;             // u2 = j + b2
    const float d1 = c * b1;
    const float d2 = c * b2;

    const float* __restrict__ xrow = x + (size_t)row * T_DIM + si;
    float* __restrict__ orow       = out + (size_t)row * L_DIM;

    const int tid = threadIdx.x;

    // Warm the streamed slice while the window constants are computed
    // (lowers to global_prefetch_b8 on gfx1250).
    __builtin_prefetch(xrow + tid, 0, 0);
    __builtin_prefetch(xrow + tid + 1024, 0, 0);

#pragma unroll
    for (int i = 0; i < 4; ++i) {
        const int   j  = tid + i * 512;             // dense across lanes: 128B/wave/instr
        const float jf = (float)j;

        // g(z) = sigmoid(softplus(z)) = 1 - 1/(2 + e^z); overflow-safe:
        // exp2 -> inf -> rcp(inf) = 0, the correct limit.
        const float e1 = __builtin_amdgcn_exp2f(__builtin_fmaf(c, jf, d1));
        const float e2 = __builtin_amdgcn_exp2f(__builtin_fmaf(c, jf, d2));
        const float r1 = __builtin_amdgcn_rcpf(2.0f + e1);
        const float r2 = __builtin_amdgcn_rcpf(2.0f + e2);
        const float w  = 2.0f * (r2 - r1);          // = 2*(g(kk*u1) - g(kk*u2))

        orow[j] = xrow[j] * w;
    }
}

extern "C" void kernel_launch(void* const* d_in, const int* in_sizes, int n_in,
                              void* d_out, int out_size, void* d_ws, size_t ws_size,
                              hipStream_t stream)
{
    (void)in_sizes; (void)n_in; (void)out_size; (void)d_ws; (void)ws_size;

    const float* x  = (const float*)d_in[0];  // (K, T) float32
    const float* kp = (const float*)d_in[1];  // (K,)   float32
    const float* si = (const float*)d_in[2];  // (K,)   float32
    float* out      = (float*)d_out;          // (K, L) float32

    dim3 grid(K_DIM);
    dim3 block(512);  // 16 wave32 waves; each thread handles 4 strided elements
    hipLaunchKernelGGL(sigdiff_window_slice_kernel, grid, block, 0, stream,
                       x, kp, si, out);
}